// GFR_81423989997578
// MI455X (gfx1250) — compile-verified
//
#include <hip/hip_runtime.h>
#include <stdint.h>
#include <stddef.h>

#ifndef __has_builtin
#define __has_builtin(x) 0
#endif

#define GFR_BLOCK 256
#define GFR_DEPTH 16  // async prefetch ring depth (timesteps ahead)

#if __has_builtin(__builtin_amdgcn_global_load_async_to_lds_b32)
#define GFR_ASYNC 1
#else
#define GFR_ASYNC 0
#endif

#if __has_builtin(__builtin_amdgcn_s_wait_asynccnt)
#define GFR_WAIT_ASYNC(n) __builtin_amdgcn_s_wait_asynccnt(n)
#else
#define GFR_WAIT_ASYNC(n) asm volatile("s_wait_asynccnt %0" ::"i"(n) : "memory")
#endif

// DS ops retire in order: allowing <=1 outstanding proves every DS read except
// the newest has completed (guards LDS slot reuse against the async refill).
#define GFR_WAIT_DS1() asm volatile("s_wait_dscnt 1" ::: "memory")

__device__ __forceinline__ float gfr_tanh(float x) {
#if __has_builtin(__builtin_amdgcn_tanhf)
  return __builtin_amdgcn_tanhf(x);          // v_tanh_f32 (CDNA5 transcendental)
#else
  // tanh(x) = 1 - 2/(exp2(2x*log2 e)+1); clamp avoids inf/inf
  float xc = fminf(fmaxf(x, -15.0f), 15.0f);
  float e  = __builtin_amdgcn_exp2f(xc * 2.8853900817779268f);
  return 1.0f - 2.0f * __builtin_amdgcn_rcpf(e + 1.0f);
#endif
}

__global__ __launch_bounds__(GFR_BLOCK) void GFR_scan_kernel(
    const float* __restrict__ Is,       // [T, B] time-major
    const float* __restrict__ a,        // [4]
    const float* __restrict__ b_gfr,    // [4]
    const float* __restrict__ dsv,      // [4]
    const float* __restrict__ pc,       // [4] poly coeffs (will be squared)
    const float* __restrict__ b_g,      // [1]
    const float* __restrict__ max_cur,  // [1]
    const float* __restrict__ max_fr,   // [1]
    float* __restrict__ out,            // [T, B]
    int T, int B) {
  const int tid = (int)threadIdx.x;
  const int gid = (int)(blockIdx.x * GFR_BLOCK) + tid;   // batch element

  // ---- uniform scalars (SMEM loads), folded constants ----
  const float a0 = a[0], a1 = a[1], a2 = a[2], a3 = a[3];
  const float d0 = 1.0f - dsv[0], d1 = 1.0f - dsv[1];
  const float d2 = 1.0f - dsv[2], d3 = 1.0f - dsv[3];
  const float f0 = 1000.0f * b_gfr[0], f1 = 1000.0f * b_gfr[1];
  const float f2 = 1000.0f * b_gfr[2], f3 = 1000.0f * b_gfr[3];
  const float c0 = pc[0] * pc[0], c1 = pc[1] * pc[1];
  const float c2 = pc[2] * pc[2], c3 = pc[3] * pc[3];
  const float inv_mc = 1.0f / max_cur[0];
  const float mfr = max_fr[0];
  const float q   = 0.25f * inv_mc;           // mean + 1/max_current folded
  const float A4  = (a0 + a1 + a2 + a3) * q;  // cur -> x contribution
  const float F4  = (f0 + f1 + f2 + f3) * q;  // fs  -> x contribution (critical FMA)
  const float nbg = -b_g[0] * inv_mc;

  float v0 = 0.0f, v1 = 0.0f, v2 = 0.0f, v3 = 0.0f, fs = 0.0f;

  auto step = [&](int t, float cur) {
    // off-critical-path: decayed state and fs-independent part of x
    const float dv0 = d0 * v0, dv1 = d1 * v1, dv2 = d2 * v2, dv3 = d3 * v3;
    const float S   = (dv0 + dv1) + (dv2 + dv3);
    const float P   = fmaf(S, q, fmaf(cur, A4, nbg));
    // critical path: fs -> x -> poly (Estrin) -> tanh -> fs
    const float x   = fmaf(fs, F4, P);
    const float x2  = x * x;
    const float p01 = fmaf(c1, x, c0);
    const float p23 = fmaf(c3, x, c2);
    const float pol = fmaf(x2, p23, p01);
    const float nfs = fmaxf(mfr * gfr_tanh(pol), 0.0f);
    // state update uses OLD fs (side path feeding next step's S)
    const float w = fs;
    v0 = fmaf(cur, a0, fmaf(w, f0, dv0));
    v1 = fmaf(cur, a1, fmaf(w, f1, dv1));
    v2 = fmaf(cur, a2, fmaf(w, f2, dv2));
    v3 = fmaf(cur, a3, fmaf(w, f3, dv3));
    fs = nfs;
    __builtin_nontemporal_store(nfs, out + (size_t)t * B + gid);
  };

#if GFR_ASYNC
  __shared__ float sbuf[GFR_DEPTH][GFR_BLOCK];
  typedef __attribute__((address_space(1))) int GI;  // b32 builtin wants int*
  typedef __attribute__((address_space(3))) int LI;
  const float* gbase = Is + gid;
#pragma unroll
  for (int k = 0; k < GFR_DEPTH; ++k) {
    __builtin_amdgcn_global_load_async_to_lds_b32(
        (GI*)(gbase + (size_t)k * B), (LI*)&sbuf[k][tid], 0, 0);
  }

  // Prologue: pull step 0's value into a register one iteration ahead.
  GFR_WAIT_ASYNC(GFR_DEPTH - 1);             // load 0 complete
  float rcur = sbuf[0][tid];

  int t = 0;
  const int Tmain = T - GFR_DEPTH;
#pragma unroll 2
  for (; t < Tmain; ++t) {
    const float cur = rcur;                  // value for step t (read last iter)
    GFR_WAIT_ASYNC(GFR_DEPTH - 2);           // load t+1 complete
    rcur = sbuf[(t + 1) & (GFR_DEPTH - 1)][tid];  // DS latency hidden by step()
    GFR_WAIT_DS1();                          // slot-t read (last iter) retired
    __builtin_amdgcn_global_load_async_to_lds_b32(
        (GI*)(gbase + (size_t)(t + GFR_DEPTH) * B),
        (LI*)&sbuf[t & (GFR_DEPTH - 1)][tid], 0, 0);
    step(t, cur);
  }
  GFR_WAIT_ASYNC(0);                         // drain tail prefetches
  for (; t < T; ++t) {
    const float cur = rcur;
    rcur = sbuf[(t + 1) & (GFR_DEPTH - 1)][tid];  // stale at t==T-1, unused
    step(t, cur);
  }
#else
  for (int t = 0; t < T; ++t) {
    float cur = __builtin_nontemporal_load(Is + (size_t)t * B + gid);
    step(t, cur);
  }
#endif
}

extern "C" void kernel_launch(void* const* d_in, const int* in_sizes, int n_in,
                              void* d_out, int out_size, void* d_ws, size_t ws_size,
                              hipStream_t stream) {
  (void)n_in; (void)out_size; (void)d_ws; (void)ws_size;
  const float* Is      = (const float*)d_in[0];
  const float* a       = (const float*)d_in[1];
  const float* b_gfr   = (const float*)d_in[2];
  const float* dsv     = (const float*)d_in[3];
  const float* pc      = (const float*)d_in[4];
  const float* b_g     = (const float*)d_in[5];
  const float* max_cur = (const float*)d_in[6];
  const float* max_fr  = (const float*)d_in[7];

  const int B = 8192;                 // fixed by problem spec
  const int T = in_sizes[0] / B;      // 4096

  dim3 grid((unsigned)(B / GFR_BLOCK));
  dim3 block(GFR_BLOCK);
  GFR_scan_kernel<<<grid, block, 0, stream>>>(Is, a, b_gfr, dsv, pc, b_g,
                                              max_cur, max_fr, (float*)d_out,
                                              T, B);
}